// TorchWrithe_62723702391611
// MI455X (gfx1250) — compile-verified
//
#include <hip/hip_runtime.h>
#include <math.h>

#define N_ATOMS 512
#define BATCH   32
#define NSEG    129795   // 509*510/2 valid (i,j) bond pairs
#define TPB     256

struct f3 { float x, y, z; };

__device__ __forceinline__ f3 sub3(f3 a, f3 b) { return {a.x - b.x, a.y - b.y, a.z - b.z}; }
__device__ __forceinline__ f3 cross3(f3 a, f3 b) {
    return {a.y * b.z - a.z * b.y, a.z * b.x - a.x * b.z, a.x * b.y - a.y * b.x};
}
__device__ __forceinline__ float dot3(f3 a, f3 b) { return a.x * b.x + a.y * b.y + a.z * b.z; }
__device__ __forceinline__ f3 nrm3(f3 a) {
    float r = rsqrtf(dot3(a, a));
    return {a.x * r, a.y * r, a.z * r};
}
__device__ __forceinline__ float clamp1(float v) { return fminf(1.0f, fmaxf(-1.0f, v)); }

// ---------------------------------------------------------------------------
// Kernel 1: zero the 32MB output (harness poisons d_out; untouched entries
// of the writhe matrix must be 0).
// ---------------------------------------------------------------------------
__global__ void zero_out_kernel(float4* __restrict__ p, int n4) {
    int t = blockIdx.x * blockDim.x + threadIdx.x;
    if (t < n4) p[t] = float4{0.0f, 0.0f, 0.0f, 0.0f};
}

// ---------------------------------------------------------------------------
// Kernel 2: writhe. blockIdx.y = batch, blockIdx.x tiles the 129795 segments.
// Stage the batch's 512x3 coordinate table (6KB) into LDS with CDNA5 async
// global->LDS loads (ASYNCcnt), then each thread computes one wr(i,j) and
// scatters it to its (statically known, conflict-free) output slots.
// ---------------------------------------------------------------------------
__global__ __launch_bounds__(TPB) void writhe_kernel(const float* __restrict__ x,
                                                     float* __restrict__ out) {
    __shared__ float sx[N_ATOMS * 3];   // 6144 bytes

    const int b = blockIdx.y;
    const float* gx = x + (size_t)b * (N_ATOMS * 3);

    // Async stage: 1536 dwords, 6 per thread. VDST = LDS byte address (low 32
    // bits of the flat shared pointer), VADDR = 64-bit global address.
    for (int k = threadIdx.x; k < N_ATOMS * 3; k += TPB) {
        unsigned lds_addr = (unsigned)(unsigned long long)(&sx[k]);
        unsigned long long gaddr = (unsigned long long)(gx + k);
        asm volatile("global_load_async_to_lds_b32 %0, %1, off"
                     :: "v"(lds_addr), "v"(gaddr)
                     : "memory");
    }
    asm volatile("s_wait_asynccnt 0" ::: "memory");
    __syncthreads();

    int t = blockIdx.x * TPB + threadIdx.x;
    if (t >= NSEG) return;

    // Invert flat index t -> (i, j): base(i) = i*(1019-i)/2, j = i+2+(t-base(i)).
    int i = (int)((1019.0f - sqrtf(1038361.0f - 8.0f * (float)t)) * 0.5f);
    if (i < 0) i = 0;
    if (i > 508) i = 508;
    while ((i + 1) * (1019 - (i + 1)) / 2 <= t) ++i;
    while (i * (1019 - i) / 2 > t) --i;
    const int j = i + 2 + (t - i * (1019 - i) / 2);

    // 4 atom positions from LDS.
    f3 p0 = {sx[3 * i],     sx[3 * i + 1], sx[3 * i + 2]};
    f3 p1 = {sx[3 * i + 3], sx[3 * i + 4], sx[3 * i + 5]};
    f3 p2 = {sx[3 * j],     sx[3 * j + 1], sx[3 * j + 2]};
    f3 p3 = {sx[3 * j + 3], sx[3 * j + 4], sx[3 * j + 5]};

    // Normalized displacements: d0=r02, d1=r03, d2=r12, d3=r13.
    f3 d0 = nrm3(sub3(p2, p0));
    f3 d1 = nrm3(sub3(p3, p0));
    f3 d2 = nrm3(sub3(p2, p1));
    f3 d3 = nrm3(sub3(p3, p1));

    // Normalized face normals of the tetrahedral cycle.
    f3 c0 = nrm3(cross3(d0, d1));
    f3 c1 = nrm3(cross3(d1, d3));
    f3 c2 = nrm3(cross3(d3, d2));
    f3 c3 = nrm3(cross3(d2, d0));

    float omega = asinf(clamp1(dot3(c0, c1)))
                + asinf(clamp1(dot3(c1, c2)))
                + asinf(clamp1(dot3(c2, c3)))
                + asinf(clamp1(dot3(c3, c0)));

    // sign((n(p3-p2) x n(p1-p0)) . d0): positive scalings don't change sign,
    // so skip the two normalizations.
    float trip = dot3(cross3(sub3(p3, p2), sub3(p1, p0)), d0);
    float sgn = (trip > 0.0f) ? 1.0f : ((trip < 0.0f) ? -1.0f : 0.0f);

    float wr = omega * sgn * 0.15915494309189535f;   // /(2*pi)

    // Resolved scatter: adj[i+1][j+1] = wr always wins; row 0 only from i==0.
    float* ob = out + (size_t)b * (N_ATOMS * N_ATOMS);
    const int r = i + 1, c = j + 1;
    ob[r * N_ATOMS + c] = wr;
    ob[c * N_ATOMS + r] = wr;
    if (i == 0) {
        ob[j] = wr;                 // adj[0][j]
        ob[j * N_ATOMS] = wr;       // adj[j][0]
    }
}

extern "C" void kernel_launch(void* const* d_in, const int* in_sizes, int n_in,
                              void* d_out, int out_size, void* d_ws, size_t ws_size,
                              hipStream_t stream) {
    const float* x = (const float*)d_in[0];   // (32, 512, 3) float32
    // d_in[1] = segments: not needed, (i,j) enumeration is closed-form.
    float* out = (float*)d_out;               // (32, 512, 512) float32

    const int n4 = out_size / 4;              // 2,097,152 float4s
    zero_out_kernel<<<(n4 + TPB - 1) / TPB, TPB, 0, stream>>>((float4*)out, n4);

    dim3 grid((NSEG + TPB - 1) / TPB, BATCH);
    writhe_kernel<<<grid, TPB, 0, stream>>>(x, out);
}